// HyperbolicGAT_19301583028951
// MI455X (gfx1250) — compile-verified
//
#include <hip/hip_runtime.h>
#include <hip/hip_bf16.h>
#include <math.h>

#define D 128
#define WARP 32

typedef float v2f __attribute__((ext_vector_type(2)));
typedef float v8f __attribute__((ext_vector_type(8)));

// ---------- helpers ----------
__device__ __forceinline__ unsigned fkey(float f) {
  unsigned u = __float_as_uint(f);
  return (u >> 31) ? ~u : (u | 0x80000000u);   // monotone order-preserving key
}
__device__ __forceinline__ float funkey(unsigned k) {
  return (k >> 31) ? __uint_as_float(k & 0x7FFFFFFFu) : __uint_as_float(~k);
}
__device__ __forceinline__ float leaky02(float v) { return v >= 0.f ? v : 0.2f * v; }
__device__ __forceinline__ float wred(float v) {
#pragma unroll
  for (int o = 16; o > 0; o >>= 1) v += __shfl_xor(v, o, WARP);
  return v;
}

// ---------- expmap0: out = tanh(||x||) * x / max(||x||,1e-15), warp/node ----------
__global__ __launch_bounds__(256) void k_expmap(const float* __restrict__ x,
                                                float* __restrict__ out, int n) {
  int node = (blockIdx.x * blockDim.x + threadIdx.x) >> 5;
  int lane = threadIdx.x & 31;
  if (node >= n) return;
  float4 v = ((const float4*)(x + (size_t)node * D))[lane];
  float ss = v.x * v.x + v.y * v.y + v.z * v.z + v.w * v.w;
  ss = wred(ss);
  float nn = fmaxf(sqrtf(ss), 1e-15f);
  float sc = tanhf(nn) / nn;
  float4 r = make_float4(v.x * sc, v.y * sc, v.z * sc, v.w * sc);
  ((float4*)(out + (size_t)node * D))[lane] = r;
}

// ---------- H = X @ W via V_WMMA_F32_16X16X4_F32 (fp32-exact) ----------
// one wave -> one 16x16 tile of H; 8 waves/block -> 16 rows x 128 cols panel.
__global__ __launch_bounds__(256) void k_gemm128(const float* __restrict__ X,
                                                 const float* __restrict__ W,
                                                 float* __restrict__ H) {
  int wave = threadIdx.x >> 5;        // 0..7 -> N tile
  int lane = threadIdx.x & 31;
  int half = lane >> 4;               // 0: K=0,1  1: K=2,3
  int l15  = lane & 15;
  int m0 = blockIdx.x * 16;
  int n0 = wave * 16;

  const float* xrow = X + (size_t)(m0 + l15) * D;
  v8f acc = {};
#pragma unroll 4
  for (int k0 = 0; k0 < D; k0 += 4) {
    int ka = k0 + half * 2;
    v2f a, b;
    a.x = xrow[ka];
    a.y = xrow[ka + 1];
    b.x = W[(size_t)ka * D + n0 + l15];
    b.y = W[(size_t)(ka + 1) * D + n0 + l15];
    acc = __builtin_amdgcn_wmma_f32_16x16x4_f32(
        /*neg_a=*/false, a, /*neg_b=*/false, b,
        /*c_mod=*/(short)0, acc, /*reuse_a=*/false, /*reuse_b=*/false);
  }
#pragma unroll
  for (int r = 0; r < 8; ++r) {
    int m = m0 + r + half * 8;        // C/D layout: vgpr r -> M=r (lanes 0-15), M=r+8 (16-31)
    H[(size_t)m * D + n0 + l15] = acc[r];
  }
}

// ---------- s = H.a_src, d = H.a_dst (warp per node) ----------
__global__ __launch_bounds__(256) void k_sd(const float* __restrict__ H,
                                            const float* __restrict__ as,
                                            const float* __restrict__ ad,
                                            float* __restrict__ sv,
                                            float* __restrict__ dv, int n) {
  int node = (blockIdx.x * blockDim.x + threadIdx.x) >> 5;
  int lane = threadIdx.x & 31;
  if (node >= n) return;
  float4 h  = ((const float4*)(H + (size_t)node * D))[lane];
  float4 a1 = ((const float4*)as)[lane];
  float4 a2 = ((const float4*)ad)[lane];
  float s = h.x * a1.x + h.y * a1.y + h.z * a1.z + h.w * a1.w;
  float d = h.x * a2.x + h.y * a2.y + h.z * a2.z + h.w * a2.w;
  s = wred(s);
  d = wred(d);
  if (lane == 0) { sv[node] = s; dv[node] = d; }
}

// ---------- init: acc[i,:] = b, mkey = 0 (== -inf key), den = 0 ----------
__global__ __launch_bounds__(256) void k_init(float* __restrict__ acc,
                                              const float* __restrict__ b,
                                              unsigned* __restrict__ mkey,
                                              float* __restrict__ den, int n) {
  int i = blockIdx.x * blockDim.x + threadIdx.x;
  if (i >= n * D) return;
  int node = i >> 7, j = i & (D - 1);
  acc[i] = b[j];
  if (j == 0) { mkey[node] = 0u; den[node] = 0.f; }
}

// ---------- pass 1: segment max of leakyrelu(s[src]+d[dst]) over dst ----------
__global__ __launch_bounds__(256) void k_edge_max(const int* __restrict__ src,
                                                  const int* __restrict__ dst,
                                                  const float* __restrict__ sv,
                                                  const float* __restrict__ dv,
                                                  unsigned* __restrict__ mkey,
                                                  int E, int Etot) {
  int e = blockIdx.x * blockDim.x + threadIdx.x;
  if (e >= Etot) return;
  int s_, d_;
  if (e < E) { s_ = src[e]; d_ = dst[e]; } else { s_ = d_ = e - E; }  // self loop
  float v = leaky02(sv[s_] + dv[d_]);
  atomicMax(&mkey[d_], fkey(v));
}

// ---------- pass 2: num = exp(e - m[dst]); den[dst] += num ----------
__global__ __launch_bounds__(256) void k_edge_exp(const int* __restrict__ src,
                                                  const int* __restrict__ dst,
                                                  const float* __restrict__ sv,
                                                  const float* __restrict__ dv,
                                                  const unsigned* __restrict__ mkey,
                                                  float* __restrict__ num,
                                                  float* __restrict__ den,
                                                  int E, int Etot) {
  int e = blockIdx.x * blockDim.x + threadIdx.x;
  if (e >= Etot) return;
  int s_, d_;
  if (e < E) { s_ = src[e]; d_ = dst[e]; } else { s_ = d_ = e - E; }
  float v = leaky02(sv[s_] + dv[d_]);
  float nu = expf(v - funkey(mkey[d_]));
  num[e] = nu;
  atomicAdd(&den[d_], nu);
}

// ---------- pass 3: acc[dst,:] += (num/den[dst]) * H[src,:]  (wave per edge) ----------
__global__ __launch_bounds__(256) void k_edge_accum(const int* __restrict__ src,
                                                    const int* __restrict__ dst,
                                                    const float* __restrict__ num,
                                                    const float* __restrict__ den,
                                                    const float* __restrict__ H,
                                                    float* __restrict__ acc,
                                                    int E, int Etot) {
  int e = (blockIdx.x * blockDim.x + threadIdx.x) >> 5;
  int lane = threadIdx.x & 31;
  if (e >= Etot) return;
  int s_, d_;
  if (e < E) { s_ = src[e]; d_ = dst[e]; } else { s_ = d_ = e - E; }
  float alpha = num[e] / den[d_];
  float4 h = ((const float4*)(H + (size_t)s_ * D))[lane];
  float* po = acc + (size_t)d_ * D + lane * 4;
  atomicAdd(po + 0, alpha * h.x);
  atomicAdd(po + 1, alpha * h.y);
  atomicAdd(po + 2, alpha * h.z);
  atomicAdd(po + 3, alpha * h.w);
}

// ---------- activation between layers: x = 2*tanh(acc) ----------
__global__ __launch_bounds__(256) void k_act(const float* __restrict__ acc,
                                             float* __restrict__ xo, int total) {
  int i = blockIdx.x * blockDim.x + threadIdx.x;
  if (i < total) xo[i] = 2.f * tanhf(acc[i]);
}

// ---------- orchestration ----------
static void run_layer(const float* Xin, const float* W, const float* as,
                      const float* ad, const float* b, float* H, float* accum,
                      float* sv, float* dv, unsigned* mkey, float* den, float* num,
                      const int* src, const int* dst, int N, int E, int Etot,
                      hipStream_t stream) {
  int nTiles = N / 16;
  k_gemm128<<<nTiles, 256, 0, stream>>>(Xin, W, H);
  k_sd<<<(N * WARP + 255) / 256, 256, 0, stream>>>(H, as, ad, sv, dv, N);
  k_init<<<(N * D + 255) / 256, 256, 0, stream>>>(accum, b, mkey, den, N);
  k_edge_max<<<(Etot + 255) / 256, 256, 0, stream>>>(src, dst, sv, dv, mkey, E, Etot);
  k_edge_exp<<<(Etot + 255) / 256, 256, 0, stream>>>(src, dst, sv, dv, mkey, num, den, E, Etot);
  k_edge_accum<<<(int)(((size_t)Etot * WARP + 255) / 256), 256, 0, stream>>>(
      src, dst, num, den, H, accum, E, Etot);
}

extern "C" void kernel_launch(void* const* d_in, const int* in_sizes, int n_in,
                              void* d_out, int out_size, void* d_ws, size_t ws_size,
                              hipStream_t stream) {
  const float* x   = (const float*)d_in[0];
  const int*   ei  = (const int*)d_in[1];
  const float* W1  = (const float*)d_in[2];
  const float* a1s = (const float*)d_in[3];
  const float* a1d = (const float*)d_in[4];
  const float* b1  = (const float*)d_in[5];
  const float* W2  = (const float*)d_in[6];
  const float* a2s = (const float*)d_in[7];
  const float* a2d = (const float*)d_in[8];
  const float* b2  = (const float*)d_in[9];
  const float* W3  = (const float*)d_in[10];
  const float* a3s = (const float*)d_in[11];
  const float* a3d = (const float*)d_in[12];
  const float* b3  = (const float*)d_in[13];

  const int N = in_sizes[0] / D;       // 50000 (divisible by 16)
  const int E = in_sizes[1] / 2;       // 1,600,000
  const int Etot = E + N;              // + self loops
  const int* src = ei;
  const int* dst = ei + E;

  // workspace carve-out
  auto al = [](size_t v) { return (v + 255) & ~(size_t)255; };
  char* w = (char*)d_ws;
  size_t nf = (size_t)N * D * sizeof(float);
  float*    XA   = (float*)w;    w += al(nf);
  float*    H    = (float*)w;    w += al(nf);
  float*    ACC  = (float*)w;    w += al(nf);
  float*    sv   = (float*)w;    w += al((size_t)N * 4);
  float*    dv   = (float*)w;    w += al((size_t)N * 4);
  unsigned* mkey = (unsigned*)w; w += al((size_t)N * 4);
  float*    den  = (float*)w;    w += al((size_t)N * 4);
  float*    num  = (float*)w;    w += al((size_t)Etot * 4);
  (void)ws_size; (void)n_in; (void)out_size;

  // h0 = expmap0(x)
  k_expmap<<<(N * WARP + 255) / 256, 256, 0, stream>>>(x, XA, N);

  // layer 1: acc = GAT(XA); XA = 2*tanh(acc)
  run_layer(XA, W1, a1s, a1d, b1, H, ACC, sv, dv, mkey, den, num, src, dst, N, E, Etot, stream);
  k_act<<<(N * D + 255) / 256, 256, 0, stream>>>(ACC, XA, N * D);

  // layer 2
  run_layer(XA, W2, a2s, a2d, b2, H, ACC, sv, dv, mkey, den, num, src, dst, N, E, Etot, stream);
  k_act<<<(N * D + 255) / 256, 256, 0, stream>>>(ACC, XA, N * D);

  // layer 3: accumulate straight into d_out (fp32 output), no activation, no copy
  run_layer(XA, W3, a3s, a3d, b3, H, (float*)d_out, sv, dv, mkey, den, num,
            src, dst, N, E, Etot, stream);
}